// CombinedLoss_8701603742379
// MI455X (gfx1250) — compile-verified
//
#include <hip/hip_runtime.h>
#include <hip/hip_bf16.h>
#include <math.h>

typedef __attribute__((ext_vector_type(2))) float v2f;
typedef __attribute__((ext_vector_type(8))) float v8f;

#define NPTS   4096      // flattened points per cloud (4*1024)
#define BATCH  4
#define NB     1024      // points per batch
#define NTILE  256       // 4096/16 tiles per dim
#define EPSV   0.005f
#define INV_EPS 200.0f
#define LOGA   (-6.93147180559945f)   // -log(1024)
#define LAB    (-13.8629436111989f)   // loga + logb
#define THRESH 1e-3f
#define MAXIT  1000

// ---------------- pack points into float4 (x,y,z,0) + squared norms ----------
__global__ void pack_kernel(const float* __restrict__ src, float4* __restrict__ dst,
                            float* __restrict__ nrm) {
    int i = blockIdx.x * blockDim.x + threadIdx.x;
    if (i >= NPTS) return;
    float x = src[3 * i + 0], y = src[3 * i + 1], z = src[3 * i + 2];
    dst[i] = make_float4(x, y, z, 0.0f);
    nrm[i] = x * x + y * y + z * z;
}

// ---------------- init mins / f / g / scalars --------------------------------
__global__ void init_kernel(unsigned* rmA, unsigned* cmA, unsigned* rmB, unsigned* cmB,
                            float* f, float* g, unsigned* sc) {
    int i = blockIdx.x * blockDim.x + threadIdx.x;
    if (i < NPTS) {
        rmA[i] = 0x7F800000u; cmA[i] = 0x7F800000u;   // +inf
        rmB[i] = 0x7F800000u; cmB[i] = 0x7F800000u;
        f[i] = 0.0f; g[i] = 0.0f;
    }
    if (i == 0) { sc[0] = 0u; sc[1] = 0u; sc[2] = 0u; }  // err_bits, it, done
}

// ---------------- WMMA tiled distance kernel ---------------------------------
// One wave per 16x16 tile of the 4096x4096 distance matrix.
// G = X·Yt via V_WMMA_F32_16X16X4_F32 (K padded 3->4), d2 = |x|^2+|y|^2-2G.
// Updates row/col mins (chamfer); on diagonal batch blocks writes the Sinkhorn
// cost matrix C[b][n][m]=sqrt(d2+1e-12) and its transpose Ct[b][m][n].
__global__ void tile_kernel(const float4* __restrict__ A4, const float* __restrict__ nA,
                            const float4* __restrict__ Y4, const float* __restrict__ nY,
                            unsigned* __restrict__ rowmin, unsigned* __restrict__ colmin,
                            float* __restrict__ Cmat, float* __restrict__ Ctmat) {
    int wave = (blockIdx.x << 3) + (threadIdx.x >> 5);
    int lane = threadIdx.x & 31;
    int ti = wave >> 8;          // row tile 0..255
    int tj = wave & 255;         // col tile 0..255
    int i0 = ti << 4, j0 = tj << 4;
    int hf  = lane >> 4;         // half of the wave selects K pair {0,1} vs {2,3}
    int l15 = lane & 15;

    float4 ca = A4[i0 + l15];
    float4 cy = Y4[j0 + l15];
    v2f a, b;
    a[0] = hf ? ca.z : ca.x;  a[1] = hf ? ca.w : ca.y;
    b[0] = hf ? cy.z : cy.x;  b[1] = hf ? cy.w : cy.y;

    v8f c = {};
    v8f d = __builtin_amdgcn_wmma_f32_16x16x4_f32(false, a, false, b,
                                                  (short)0, c, false, false);

    int j = j0 + l15;
    float nyj = nY[j];
    float colLocal = INFINITY;
    float d2v[8];
#pragma unroll
    for (int v = 0; v < 8; ++v) {
        int i = i0 + (hf << 3) + v;
        float d2 = fmaxf(nA[i] + nyj - 2.0f * d[v], 0.0f);
        d2v[v] = d2;
        colLocal = fminf(colLocal, d2);
        float r = d2;                                   // row min over 16 lanes of this half
        r = fminf(r, __shfl_xor(r, 1, 32));
        r = fminf(r, __shfl_xor(r, 2, 32));
        r = fminf(r, __shfl_xor(r, 4, 32));
        r = fminf(r, __shfl_xor(r, 8, 32));
        if (l15 == 0) atomicMin(&rowmin[i], __float_as_uint(r));
    }
    colLocal = fminf(colLocal, __shfl_xor(colLocal, 16, 32));
    if (hf == 0) atomicMin(&colmin[j], __float_as_uint(colLocal));

    if (Cmat != nullptr && (ti >> 6) == (tj >> 6)) {    // diagonal batch block
        int bb = ti >> 6;
        int m = j & (NB - 1);
        int nbase = (i0 & (NB - 1)) + (hf << 3);
#pragma unroll
        for (int v = 0; v < 8; ++v) {
            int n = nbase + v;
            float cc = sqrtf(d2v[v] + 1e-12f);
            Cmat [(((bb << 10) + n) << 10) + m] = cc;
            Ctmat[(((bb << 10) + m) << 10) + n] = cc;
        }
    }
}

// ---------------- Sinkhorn: streaming logsumexp helpers ----------------------
__device__ __forceinline__ void lse_push(float s, float& M, float& S) {
    if (s > M) { S = S * __expf(M - s) + 1.0f; M = s; }
    else       { S += __expf(s - M); }
}
__device__ __forceinline__ void lse_reduce(float& M, float& S) {
#pragma unroll
    for (int mask = 16; mask > 0; mask >>= 1) {
        float oM = __shfl_xor(M, mask, 32);
        float oS = __shfl_xor(S, mask, 32);
        float nM = fmaxf(M, oM);
        S = S * __expf(M - nM) + oS * __expf(oM - nM);
        M = nM;
    }
}

// g_new[b,m] = -eps*(LSE_n((f[n]-C[n,m])/eps) + loga) ; one wave per (b,m)
__global__ void g_kernel(const float* __restrict__ Ct, const float* __restrict__ f,
                         float* __restrict__ g, const unsigned* __restrict__ sc) {
    if (sc[2]) return;
    int wave = (blockIdx.x << 3) + (threadIdx.x >> 5);
    int lane = threadIdx.x & 31;
    int bb = wave >> 10, m = wave & (NB - 1);
    const float* crow = Ct + ((((size_t)bb << 10) + m) << 10);
    const float* frow = f + (bb << 10);
    float M = -INFINITY, S = 0.0f;
#pragma unroll 4
    for (int n = lane; n < NB; n += 32)
        lse_push((frow[n] - crow[n]) * INV_EPS, M, S);
    lse_reduce(M, S);
    if (lane == 0) g[(bb << 10) + m] = -EPSV * (M + __logf(S) + LOGA);
}

// f_new[b,n] = -eps*(LSE_m((g[m]-C[n,m])/eps) + logb) ; tracks max |f_new-f|
__global__ void f_kernel(const float* __restrict__ C, float* __restrict__ f,
                         const float* __restrict__ g, unsigned* __restrict__ sc) {
    if (sc[2]) return;
    int wave = (blockIdx.x << 3) + (threadIdx.x >> 5);
    int lane = threadIdx.x & 31;
    int bb = wave >> 10, n = wave & (NB - 1);
    const float* crow = C + ((((size_t)bb << 10) + n) << 10);
    const float* grow = g + (bb << 10);
    float M = -INFINITY, S = 0.0f;
#pragma unroll 4
    for (int m = lane; m < NB; m += 32)
        lse_push((grow[m] - crow[m]) * INV_EPS, M, S);
    lse_reduce(M, S);
    if (lane == 0) {
        float fnew = -EPSV * (M + __logf(S) + LOGA);  // logb == loga
        int idx = (bb << 10) + n;
        float diff = fabsf(fnew - f[idx]);
        f[idx] = fnew;
        atomicMax(&sc[0], __float_as_uint(diff));     // diff >= 0: bit-monotone
    }
}

// per-iteration convergence check (replicates the jax while_loop condition)
__global__ void step_kernel(unsigned* sc) {
    if (sc[2]) return;
    float err = __uint_as_float(sc[0]);
    unsigned it = sc[1] + 1u;
    sc[1] = it;
    if (it >= (unsigned)MAXIT || err <= THRESH) sc[2] = 1u;
    sc[0] = 0u;   // reset err accumulator for next iteration
}

// ---------------- emd = mean_b sum P*C, stage 1 (deterministic tree) ---------
__global__ void emd_stage1(const float* __restrict__ C, const float* __restrict__ f,
                           const float* __restrict__ g, float* __restrict__ partial) {
    __shared__ float red[256];
    size_t base = (size_t)blockIdx.x * 4096;
    float acc = 0.0f;
    for (int k = threadIdx.x; k < 4096; k += 256) {
        size_t e = base + k;
        int bb = (int)(e >> 20);
        int n  = (int)((e >> 10) & (NB - 1));
        int m  = (int)(e & (NB - 1));
        float cc = C[e];
        float p = __expf((f[(bb << 10) + n] + g[(bb << 10) + m] - cc) * INV_EPS + LAB);
        acc += p * cc;
    }
    red[threadIdx.x] = acc; __syncthreads();
    for (int s = 128; s > 0; s >>= 1) {
        if (threadIdx.x < (unsigned)s) red[threadIdx.x] += red[threadIdx.x + s];
        __syncthreads();
    }
    if (threadIdx.x == 0) partial[blockIdx.x] = red[0];
}

// ---------------- final combine: chamfers + mse + emd ------------------------
__device__ float block_sum(float v, float* red) {
    int t = threadIdx.x;
    red[t] = v; __syncthreads();
    for (int s = 128; s > 0; s >>= 1) {
        if (t < s) red[t] += red[t + s];
        __syncthreads();
    }
    float r = red[0]; __syncthreads();
    return r;
}

__global__ void combine_kernel(const float* __restrict__ pc_d, const float* __restrict__ pc2,
                               const unsigned* __restrict__ rmA, const unsigned* __restrict__ cmA,
                               const unsigned* __restrict__ rmB, const unsigned* __restrict__ cmB,
                               const float* __restrict__ emdPartial, float* __restrict__ out) {
    __shared__ float red[256];
    int t = threadIdx.x;
    float sColA = 0.f, sRowA = 0.f, sColB = 0.f, sRowB = 0.f, sMse = 0.f, sEmd = 0.f;
    for (int i = t; i < NPTS; i += 256) {
        sColA += sqrtf(__uint_as_float(cmA[i]));
        sRowA += sqrtf(__uint_as_float(rmA[i]));
        sColB += sqrtf(__uint_as_float(cmB[i]));
        sRowB += sqrtf(__uint_as_float(rmB[i]));
    }
    for (int i = t; i < NPTS * 3; i += 256) {
        float dd = pc_d[i] - pc2[i];
        sMse += dd * dd;
    }
    for (int i = t; i < 1024; i += 256) sEmd += emdPartial[i];

    float colA = block_sum(sColA, red);
    float rowA = block_sum(sRowA, red);
    float colB = block_sum(sColB, red);
    float rowB = block_sum(sRowB, red);
    float mse  = block_sum(sMse,  red);
    float emd  = block_sum(sEmd,  red);
    if (t == 0) {
        float cd_a = colA / NPTS + rowA / NPTS;
        float cd_b = colB / NPTS + rowB / NPTS;
        float mseV = mse / (NPTS * 3);
        float emdV = emd / BATCH;
        out[0] = mseV + 0.5f * cd_a + 0.5f * emdV + cd_b;
    }
}

// ---------------- launcher ---------------------------------------------------
extern "C" void kernel_launch(void* const* d_in, const int* in_sizes, int n_in,
                              void* d_out, int out_size, void* d_ws, size_t ws_size,
                              hipStream_t stream) {
    (void)in_sizes; (void)n_in; (void)out_size; (void)ws_size;
    const float* pc_a = (const float*)d_in[0];
    const float* pc_b = (const float*)d_in[1];
    const float* pc_d = (const float*)d_in[2];
    const float* pc2  = (const float*)d_in[3];
    char* ws = (char*)d_ws;

    float4*   xa4  = (float4*)(ws + 0);
    float4*   xb4  = (float4*)(ws + 65536);
    float4*   y4   = (float4*)(ws + 131072);
    float*    na   = (float*)(ws + 196608);
    float*    nb   = (float*)(ws + 212992);
    float*    ny   = (float*)(ws + 229376);
    unsigned* rmA  = (unsigned*)(ws + 245760);
    unsigned* cmA  = (unsigned*)(ws + 262144);
    unsigned* rmB  = (unsigned*)(ws + 278528);
    unsigned* cmB  = (unsigned*)(ws + 294912);
    float*    f    = (float*)(ws + 311296);
    float*    g    = (float*)(ws + 327680);
    unsigned* sc   = (unsigned*)(ws + 344064);
    float*    part = (float*)(ws + 344128);
    float*    C    = (float*)(ws + 348224);
    float*    Ct   = (float*)(ws + 348224 + 16777216);

    pack_kernel<<<16, 256, 0, stream>>>(pc_a, xa4, na);
    pack_kernel<<<16, 256, 0, stream>>>(pc_b, xb4, nb);
    pack_kernel<<<16, 256, 0, stream>>>(pc2,  y4,  ny);
    init_kernel<<<16, 256, 0, stream>>>(rmA, cmA, rmB, cmB, f, g, sc);

    // 65536 tiles, 8 waves/block
    tile_kernel<<<8192, 256, 0, stream>>>(xa4, na, y4, ny, rmA, cmA, C, Ct);
    tile_kernel<<<8192, 256, 0, stream>>>(xb4, nb, y4, ny, rmB, cmB, nullptr, nullptr);

    for (int it = 0; it < MAXIT; ++it) {
        g_kernel<<<512, 256, 0, stream>>>(Ct, f, g, sc);
        f_kernel<<<512, 256, 0, stream>>>(C, f, g, sc);
        step_kernel<<<1, 1, 0, stream>>>(sc);
    }

    emd_stage1<<<1024, 256, 0, stream>>>(C, f, g, part);
    combine_kernel<<<1, 256, 0, stream>>>(pc_d, pc2, rmA, cmA, rmB, cmB, part,
                                          (float*)d_out);
}